// SAGE_47914655154529
// MI455X (gfx1250) — compile-verified
//
#include <hip/hip_runtime.h>
#include <math.h>

#define N_NODES 100000
#define N_EDGES 1600000
#define F       128          // F_IN == EMB == 128
#define NG      128          // graphs
#define GFEAT   32
#define LDS_STRIDE 132       // 128 + 4 pad -> conflict-free row access

typedef float v2f __attribute__((ext_vector_type(2)));
typedef float v8f __attribute__((ext_vector_type(8)));

// ---- float <-> order-preserving unsigned key (for atomic max on f32) ----
__device__ __forceinline__ unsigned f2k(float f) {
    unsigned b = __float_as_uint(f);
    return (b & 0x80000000u) ? ~b : (b | 0x80000000u);
}
__device__ __forceinline__ float k2f(unsigned k) {
    unsigned b = (k & 0x80000000u) ? (k & 0x7FFFFFFFu) : ~k;
    return __uint_as_float(b);
}

// ---------------------------------------------------------------- fill zero
__global__ void fill_zero(float* __restrict__ p, int n) {
    int i = blockIdx.x * blockDim.x + threadIdx.x;
    if (i < n) p[i] = 0.0f;
}

// ------------------------------------------------------------- degree count
__global__ void count_deg(const int* __restrict__ tgt, float* __restrict__ cnt) {
    int e = blockIdx.x * blockDim.x + threadIdx.x;
    if (e < N_EDGES) atomicAdd(&cnt[tgt[e]], 1.0f);
}

// cnt -> 1/max(cnt,1) in place; also per-graph node counts
__global__ void finalize_cnt(float* __restrict__ inv, const int* __restrict__ batch,
                             float* __restrict__ gcnt) {
    int n = blockIdx.x * blockDim.x + threadIdx.x;
    if (n < N_NODES) {
        inv[n] = 1.0f / fmaxf(inv[n], 1.0f);
        atomicAdd(&gcnt[batch[n]], 1.0f);
    }
}

// ------------------------------------------------- edge scatter: agg[t] += feat[s]
// one wave32 per edge, each lane handles 4 consecutive features (float4 load)
__global__ void scatter_add(const int* __restrict__ src, const int* __restrict__ tgt,
                            const float* __restrict__ feat, float* __restrict__ agg) {
    int tid = blockIdx.x * blockDim.x + threadIdx.x;   // N_EDGES*32 threads
    int e   = tid >> 5;
    if (e >= N_EDGES) return;
    int c4  = (tid & 31) << 2;
    int s = src[e], t = tgt[e];
    const float4 v = *(const float4*)(feat + (size_t)s * F + c4);
    float* p = agg + (size_t)t * F + c4;
    atomicAdd(p + 0, v.x);
    atomicAdd(p + 1, v.y);
    atomicAdd(p + 2, v.z);
    atomicAdd(p + 3, v.w);
}

// ---------------------------------------------------------------- SAGE GEMM
// out_tile = (A1 * inv[row]) @ B1 + A2 @ B2 + bias      (K = F = 128)
// RELU_STORE: relu + store to out.   else: fused mean/max pooling epilogue.
// block = 256 threads (8 wave32); block computes rows [16) x cols [128).
template <bool RELU_STORE>
__global__ void sage_gemm(const float* __restrict__ A1, const float* __restrict__ inv,
                          const float* __restrict__ B1, const float* __restrict__ A2,
                          const float* __restrict__ B2, const float* __restrict__ bias,
                          float* __restrict__ out,
                          const int* __restrict__ batch,
                          float* __restrict__ gsum, unsigned* __restrict__ gmaxK) {
    __shared__ float As1[16 * LDS_STRIDE];
    __shared__ float As2[16 * LDS_STRIDE];

    const int tid     = threadIdx.x;
    const int tileRow = blockIdx.x << 4;               // 6250 * 16 == 100000 exactly

    // cooperative stage of both A slabs; mean-scale folded into A1 here
    for (int i = tid; i < 16 * 32; i += 256) {
        int r  = i >> 5;
        int c4 = (i & 31) << 2;
        const size_t gofs = (size_t)(tileRow + r) * F + c4;
        float s = inv[tileRow + r];
        float4 a1 = *(const float4*)(A1 + gofs);
        float4 a2 = *(const float4*)(A2 + gofs);
        float* p1 = &As1[r * LDS_STRIDE + c4];
        float* p2 = &As2[r * LDS_STRIDE + c4];
        p1[0] = a1.x * s; p1[1] = a1.y * s; p1[2] = a1.z * s; p1[3] = a1.w * s;
        p2[0] = a2.x;     p2[1] = a2.y;     p2[2] = a2.z;     p2[3] = a2.w;
    }
    __syncthreads();

    const int wave    = tid >> 5;
    const int lane    = tid & 31;
    const int l15     = lane & 15;
    const int khalf   = (lane >> 4) << 1;              // 0 for lanes 0-15, 2 for 16-31
    const int colBase = wave << 4;                     // 8 waves cover 128 cols
    const int col     = colBase + l15;
    const int arow    = l15 * LDS_STRIDE;

    v8f acc = {};
    #pragma unroll
    for (int k0 = 0; k0 < F; k0 += 4) {
        // A frag: 16x4 f32 — lane half selects K pair {k0,k0+1} or {k0+2,k0+3}
        v2f a1; a1.x = As1[arow + k0 + khalf];
                a1.y = As1[arow + k0 + khalf + 1];
        // B frag: 4x16 f32 — VGPR0 = rows {k0, k0+2}, VGPR1 = rows {k0+1, k0+3}
        v2f b1; b1.x = B1[(k0 + khalf)     * F + col];
                b1.y = B1[(k0 + khalf + 1) * F + col];
        acc = __builtin_amdgcn_wmma_f32_16x16x4_f32(false, a1, false, b1,
                                                    (short)0, acc, false, false);
        v2f a2; a2.x = As2[arow + k0 + khalf];
                a2.y = As2[arow + k0 + khalf + 1];
        v2f b2; b2.x = B2[(k0 + khalf)     * F + col];
                b2.y = B2[(k0 + khalf + 1) * F + col];
        acc = __builtin_amdgcn_wmma_f32_16x16x4_f32(false, a2, false, b2,
                                                    (short)0, acc, false, false);
    }

    const float bcol = bias[col];
    const int   rofs = (lane >> 4) << 3;               // D: lanes 16-31 hold M=j+8

    if (RELU_STORE) {
        #pragma unroll
        for (int j = 0; j < 8; ++j) {
            int row = tileRow + j + rofs;
            out[(size_t)row * F + col] = fmaxf(acc[j] + bcol, 0.0f);
        }
    } else {
        #pragma unroll
        for (int j = 0; j < 8; ++j) {
            int row = tileRow + j + rofs;
            float v = acc[j] + bcol;
            int g   = batch[row];
            atomicAdd(&gsum[g * F + col], v);
            atomicMax(&gmaxK[g * F + col], f2k(v));
        }
    }
}

// ---------------------------------------------------------------- readout
// one thread per graph: mean | max | (g_features @ W_g + b_g) -> W_o -> log_softmax
__global__ void readout(const float* __restrict__ gsum, const unsigned* __restrict__ gmaxK,
                        const float* __restrict__ gcnt, const float* __restrict__ gfeat,
                        const float* __restrict__ Wg, const float* __restrict__ bg,
                        const float* __restrict__ Wo, const float* __restrict__ bo,
                        float* __restrict__ out) {
    int g = threadIdx.x;
    if (g >= NG) return;
    float cg = fmaxf(gcnt[g], 1.0f);
    float l0 = bo[0], l1 = bo[1];
    for (int j = 0; j < F; ++j) {
        float mean = gsum[g * F + j] / cg;
        float mx   = k2f(gmaxK[g * F + j]);
        l0 += mean * Wo[j * 2 + 0] + mx * Wo[(F + j) * 2 + 0];
        l1 += mean * Wo[j * 2 + 1] + mx * Wo[(F + j) * 2 + 1];
    }
    for (int j = 0; j < F; ++j) {
        float gl = bg[j];
        for (int k = 0; k < GFEAT; ++k) gl += gfeat[g * GFEAT + k] * Wg[k * F + j];
        l0 += gl * Wo[(2 * F + j) * 2 + 0];
        l1 += gl * Wo[(2 * F + j) * 2 + 1];
    }
    float m   = fmaxf(l0, l1);
    float lse = m + logf(expf(l0 - m) + expf(l1 - m));
    out[g * 2 + 0] = l0 - lse;
    out[g * 2 + 1] = l1 - lse;
}

// ---------------------------------------------------------------- launch
extern "C" void kernel_launch(void* const* d_in, const int* in_sizes, int n_in,
                              void* d_out, int out_size, void* d_ws, size_t ws_size,
                              hipStream_t stream) {
    const float* x     = (const float*)d_in[0];
    const int*   edges = (const int*)  d_in[1];   // [2, E]: src rows 0..E-1, tgt E..2E-1
    const int*   batch = (const int*)  d_in[2];
    const float* gfeat = (const float*)d_in[3];
    const float* Wl0   = (const float*)d_in[4];
    const float* bl0   = (const float*)d_in[5];
    const float* Wr0   = (const float*)d_in[6];
    const float* Wl1   = (const float*)d_in[7];
    const float* bl1   = (const float*)d_in[8];
    const float* Wr1   = (const float*)d_in[9];
    const float* Wg    = (const float*)d_in[10];
    const float* bg    = (const float*)d_in[11];
    const float* Wo    = (const float*)d_in[12];
    const float* bo    = (const float*)d_in[13];
    float* out = (float*)d_out;

    const int* src = edges;
    const int* tgt = edges + N_EDGES;

    // workspace carve (all zero-init regions contiguous, h0 last / uninitialized)
    float*    agg   = (float*)d_ws;                            // N*F
    float*    inv   = agg + (size_t)N_NODES * F;               // N   (counts, then 1/deg)
    float*    gsum  = inv + N_NODES;                           // NG*F
    unsigned* gmaxK = (unsigned*)(gsum + NG * F);              // NG*F
    float*    gcnt  = (float*)(gmaxK + NG * F);                // NG
    float*    h0    = gcnt + NG;                               // N*F

    const int zspan = N_NODES * F + N_NODES + NG * F + NG * F + NG;
    const int nodeF = N_NODES * F;

    fill_zero<<<(zspan + 255) / 256, 256, 0, stream>>>(agg, zspan);
    count_deg<<<(N_EDGES + 255) / 256, 256, 0, stream>>>(tgt, inv);
    finalize_cnt<<<(N_NODES + 255) / 256, 256, 0, stream>>>(inv, batch, gcnt);

    // ---- layer 0: agg(x) -> h0 = relu(mean@Wl0 + x@Wr0 + b)
    scatter_add<<<(N_EDGES * 32) / 256, 256, 0, stream>>>(src, tgt, x, agg);
    sage_gemm<true><<<N_NODES / 16, 256, 0, stream>>>(agg, inv, Wl0, x, Wr0, bl0,
                                                      h0, nullptr, nullptr, nullptr);

    // ---- layer 1: agg(h0) -> h1 tiles fused directly into pooling
    fill_zero<<<(nodeF + 255) / 256, 256, 0, stream>>>(agg, nodeF);
    scatter_add<<<(N_EDGES * 32) / 256, 256, 0, stream>>>(src, tgt, h0, agg);
    sage_gemm<false><<<N_NODES / 16, 256, 0, stream>>>(agg, inv, Wl1, h0, Wr1, bl1,
                                                       nullptr, batch, gsum, gmaxK);

    // ---- readout: 128 graphs -> [128,2] log-softmax
    readout<<<1, 128, 0, stream>>>(gsum, gmaxK, gcnt, gfeat, Wg, bg, Wo, bo, out);
}